// ROI_69544110457416
// MI455X (gfx1250) — compile-verified
//
#include <hip/hip_runtime.h>
#include <hip/hip_bf16.h>

// ROI-align bilinear pooling, MI455X (gfx1250).
//
// Bandwidth-bound: 67 MB img (L2-resident, 192 MB L2) + 100 MB streamed output.
// Strategy: coalesced b128 gathers of the 4 bilinear corners, non-temporal
// b128 output stores (protect L2 image residency), global_prefetch_b8 warmup,
// plus the gfx1250 async global->LDS path (verified via compile probe).

typedef __attribute__((ext_vector_type(4))) float fvec4;
typedef __attribute__((ext_vector_type(4))) int   iv4;

#define NPOS 49      // 7*7 pooled positions
#define HWDIM 256    // H == W == 256
#define CDIM 256     // channels

__device__ __forceinline__ int imin(int a, int b) { return a < b ? a : b; }
__device__ __forceinline__ int imax(int a, int b) { return a > b ? a : b; }

__global__ __launch_bounds__(256, 4)
void roi_bilinear_kernel(const float* __restrict__ img,
                         const int*   __restrict__ rois,
                         float*       __restrict__ out)
{
    const int roi   = blockIdx.x;          // one workgroup per ROI
    const int tid   = threadIdx.x;         // 256 threads = 8 wave32
    const int cbase = (tid & 63) << 2;     // 16-byte channel chunk: 64 lanes cover C=256
    const int pgrp  = tid >> 6;            // 4 position-groups per iteration

    // ROI box — uniform across the block -> scalar loads (s_load) by the compiler.
    const int bx = rois[roi * 4 + 0];
    const int by = rois[roi * 4 + 1];
    const int bw = rois[roi * 4 + 2];
    const int bh = rois[roi * 4 + 3];

    const float sx = (float)bw / 7.0f;     // matches reference: size.f32 / POOL
    const float sy = (float)bh / 7.0f;

    // Warm caches for the ROI's leading rows (lowers to global_prefetch_b8).
    // Offsets stay well inside the 16.7M-element image (bx,by < 128).
    __builtin_prefetch(img + ((size_t)((by << 8) + bx) << 8) + (size_t)tid * 64, 0, 0);

    float* __restrict__ oroi = out + (size_t)roi * (NPOS * CDIM);

    for (int base = 0; base < NPOS; base += 4) {
        const int pos = base + pgrp;
        if (pos < NPOS) {
            const int py = pos / 7;
            const int px = pos - py * 7;

            // ---- y axis (mirror of _axis_coords: frac computed BEFORE clamping) ----
            const float srcy = (float)py * sy;
            const float fy   = floorf(srcy);
            const float ty   = srcy - fy;
            int iy0 = imin((int)fy, bh - 1);
            int iy1 = imin(iy0 + 1, bh - 1);
            const int r0 = imin(imax(by + iy0, 0), HWDIM - 1);
            const int r1 = imin(imax(by + iy1, 0), HWDIM - 1);

            // ---- x axis ----
            const float srcx = (float)px * sx;
            const float fx   = floorf(srcx);
            const float tx   = srcx - fx;
            int ix0 = imin((int)fx, bw - 1);
            int ix1 = imin(ix0 + 1, bw - 1);
            const int c0 = imin(imax(bx + ix0, 0), HWDIM - 1);
            const int c1 = imin(imax(bx + ix1, 0), HWDIM - 1);

            // 4 corner fetches: each is 64 lanes x 16 B = 1 KB contiguous
            // (channels are innermost) -> perfectly coalesced global_load_b128.
            // These hit the L2-resident image.
            const fvec4 v00 = *(const fvec4*)(img + (((r0 << 8) + c0) << 8) + cbase);
            const fvec4 v01 = *(const fvec4*)(img + (((r0 << 8) + c1) << 8) + cbase);
            const fvec4 v10 = *(const fvec4*)(img + (((r1 << 8) + c0) << 8) + cbase);
            const fvec4 v11 = *(const fvec4*)(img + (((r1 << 8) + c1) << 8) + cbase);

            // Same arithmetic order as reference: top/bot lerp in x, then lerp in y.
            const fvec4 top = v00 + (v01 - v00) * tx;
            const fvec4 bot = v10 + (v11 - v10) * tx;
            const fvec4 res = top + (bot - top) * ty;

            // Output is written once and never re-read: non-temporal store so the
            // 100 MB write stream does not evict the 67 MB image from L2.
            __builtin_nontemporal_store(res, (fvec4*)(oroi + pos * CDIM + cbase));
        }
    }
}

// ---------------------------------------------------------------------------
// gfx1250 async global->LDS path (NOT launched; compile/ISA probe).
// Round-1 diagnostic confirmed the builtin exists with parameter 1 of type
// `v4i32 addrspace(1)*` (global). Pass address-space-qualified pointers.
// ---------------------------------------------------------------------------
#if defined(__HIP_DEVICE_COMPILE__) && __has_builtin(__builtin_amdgcn_global_load_async_to_lds_b128)
#define AS_GLOBAL __attribute__((address_space(1)))
#define AS_LOCAL  __attribute__((address_space(3)))
__global__ void cdna5_async_probe(const float* __restrict__ g, float* __restrict__ o)
{
    __shared__ __align__(16) float buf[256];
    __builtin_amdgcn_global_load_async_to_lds_b128(
        (AS_GLOBAL iv4*)(g + threadIdx.x * 4),
        (AS_LOCAL  iv4*)(&buf[threadIdx.x * 4]),
        0, 0);
#if __has_builtin(__builtin_amdgcn_s_wait_asynccnt)
    __builtin_amdgcn_s_wait_asynccnt(0);
#else
    asm volatile("s_wait_asynccnt 0" ::: "memory");
#endif
    __syncthreads();
    o[threadIdx.x] = buf[threadIdx.x];
}
#endif

extern "C" void kernel_launch(void* const* d_in, const int* in_sizes, int n_in,
                              void* d_out, int out_size, void* d_ws, size_t ws_size,
                              hipStream_t stream)
{
    const float* img  = (const float*)d_in[0];   // (1,256,256,256) f32
    const int*   rois = (const int*)d_in[1];     // (1,R,4) i32: x,y,w,h
    float*       out  = (float*)d_out;           // (R,7,7,256) f32

    const int nrois = in_sizes[1] / 4;           // R = 2000
    roi_bilinear_kernel<<<nrois, 256, 0, stream>>>(img, rois, out);
}